// LengthRegulator_14637248544773
// MI455X (gfx1250) — compile-verified
//
#include <hip/hip_runtime.h>
#include <hip/hip_bf16.h>

// ---------------------------------------------------------------------------
// LengthRegulator for MI455X (gfx1250, wave32)
//   x:        (16, 1024, 384) f32
//   duration: (16, 1024)      int
//   out:      (16, 6144, 384) f32  followed by mel_len (16) stored as f32
//
// Pure data movement (~176 MB @ 23.3 TB/s => ~7.6 us floor). Each output row
// is either an exact copy of one x-row or zeros, so we use the CDNA5
// VGPR-bypassing async path:
//   global_load_async_to_lds_b128 -> s_wait_asynccnt -> global_store_async_from_lds_b128
// ---------------------------------------------------------------------------

#define NB      16
#define TPH     1024
#define DMODEL  384
#define MAXDUR  6
#define MAXLEN  (TPH * MAXDUR)   // 6144
#define RPB     8                // rows (output timesteps) per block, 1 wave32 each

#if defined(__has_builtin)
#  if __has_builtin(__builtin_amdgcn_global_load_async_to_lds_b128) && \
      __has_builtin(__builtin_amdgcn_global_store_async_from_lds_b128) && \
      __has_builtin(__builtin_amdgcn_s_wait_asynccnt)
#    define HAVE_ASYNC_LDS 1
#  endif
#endif
#ifndef HAVE_ASYNC_LDS
#  define HAVE_ASYNC_LDS 0
#endif

// 128-bit transfer element type expected by the async builtins
typedef int v4i __attribute__((vector_size(16)));
typedef __attribute__((address_space(1))) v4i GlobV4;
typedef __attribute__((address_space(3))) v4i LdsV4;

// ---------------------------------------------------------------------------
// Kernel A: per-batch inclusive scan of durations (LDS Hillis-Steele),
// scatter-expand the index map, tag tail rows with -1, emit mel_len.
// ---------------------------------------------------------------------------
__global__ __launch_bounds__(TPH)
void lr_scan_expand_kernel(const int* __restrict__ dur,
                           int* __restrict__ idx_buf,
                           float* __restrict__ mel_out)
{
    __shared__ int s[TPH];
    const int b = blockIdx.x;
    const int i = threadIdx.x;

    const int d = dur[(size_t)b * TPH + i];
    s[i] = d;
    __syncthreads();

    // inclusive scan over 1024 elements: 10 rounds
    #pragma unroll
    for (int off = 1; off < TPH; off <<= 1) {
        const int v = (i >= off) ? s[i - off] : 0;
        __syncthreads();
        s[i] += v;
        __syncthreads();
    }

    const int cum   = s[i];          // sum(dur[0..i])
    const int start = cum - d;       // exclusive prefix
    int* __restrict__ row = idx_buf + (size_t)b * MAXLEN;

    // searchsorted(cum, t, 'right') == i  for t in [start, cum)
    for (int k = start; k < cum; ++k) row[k] = i;

    const int mel = s[TPH - 1];
    // rows at/after mel_len are masked to zero in the reference
    for (int t = mel + i; t < MAXLEN; t += TPH) row[t] = -1;

    if (i == TPH - 1) mel_out[b] = (float)cum;   // cum == mel here
}

// ---------------------------------------------------------------------------
// Kernel B: one wave32 per output row. Row = 384 f32 = 1536 B = 3 x (32 lanes
// x 16 B). Copy x-row -> out-row, or zeros for masked rows, via async LDS.
// ---------------------------------------------------------------------------
__global__ __launch_bounds__(RPB * 32)
void lr_copy_kernel(const float* __restrict__ x,
                    const int*   __restrict__ idx_buf,
                    float*       __restrict__ out)
{
    const int b    = blockIdx.y;
    const int wave = threadIdx.x >> 5;
    const int lane = threadIdx.x & 31;
    const int t    = blockIdx.x * RPB + wave;

#if HAVE_ASYNC_LDS
    // region 0: 1536 B of zeros shared by all waves; regions 1..RPB: per-wave stage
    __shared__ float lds[(RPB + 1) * DMODEL];
    for (int k = threadIdx.x; k < DMODEL; k += RPB * 32) lds[k] = 0.0f;
    __syncthreads();
#endif

    const int idx = idx_buf[(size_t)b * MAXLEN + t];         // uniform per wave
    float* __restrict__ orow = out + ((size_t)b * MAXLEN + t) * DMODEL;

#if HAVE_ASYNC_LDS
    float* stage = lds + (idx >= 0 ? (wave + 1) * DMODEL : 0);
    if (idx >= 0) {
        const float* xrow = x + ((size_t)b * TPH + idx) * DMODEL;
        #pragma unroll
        for (int k = 0; k < 3; ++k) {
            __builtin_amdgcn_global_load_async_to_lds_b128(
                (GlobV4*)(xrow  + (lane + 32 * k) * 4),
                (LdsV4*) (stage + (lane + 32 * k) * 4),
                0, 0);
        }
        __builtin_amdgcn_s_wait_asynccnt(0);   // LDS data ready before store reads it
    }
    #pragma unroll
    for (int k = 0; k < 3; ++k) {
        __builtin_amdgcn_global_store_async_from_lds_b128(
            (GlobV4*)(orow  + (lane + 32 * k) * 4),
            (LdsV4*) (stage + (lane + 32 * k) * 4),
            0, 0);
    }
    // implicit wait-idle at s_endpgm covers outstanding async stores
#else
    if (idx >= 0) {
        const float* xrow = x + ((size_t)b * TPH + idx) * DMODEL;
        __builtin_prefetch(xrow + lane * 4, 0, 0);           // global_prefetch_b8
        #pragma unroll
        for (int k = 0; k < 3; ++k) {
            const float4 v = *(const float4*)(xrow + (lane + 32 * k) * 4);
            *(float4*)(orow + (lane + 32 * k) * 4) = v;
        }
    } else {
        const float4 z = make_float4(0.f, 0.f, 0.f, 0.f);
        #pragma unroll
        for (int k = 0; k < 3; ++k)
            *(float4*)(orow + (lane + 32 * k) * 4) = z;
    }
#endif
}

// ---------------------------------------------------------------------------
extern "C" void kernel_launch(void* const* d_in, const int* in_sizes, int n_in,
                              void* d_out, int out_size, void* d_ws, size_t ws_size,
                              hipStream_t stream)
{
    const float* x   = (const float*)d_in[0];
    const int*   dur = (const int*)  d_in[1];   // integer input -> const int*
    // d_in[2] = max_len scalar; compile-time constant MAXLEN used instead

    float* out     = (float*)d_out;
    float* mel_out = out + (size_t)NB * MAXLEN * DMODEL;  // tuple tail: mel_len
    int*   idx_buf = (int*)d_ws;                          // 16*6144*4 = 384 KB

    lr_scan_expand_kernel<<<NB, TPH, 0, stream>>>(dur, idx_buf, mel_out);

    dim3 grid(MAXLEN / RPB, NB);
    lr_copy_kernel<<<grid, RPB * 32, 0, stream>>>(x, idx_buf, out);
}